// DistanceWeightedMessagePassing_36094905155955
// MI455X (gfx1250) — compile-verified
//
#include <hip/hip_runtime.h>
#include <hip/hip_bf16.h>
#include <math.h>

typedef __attribute__((ext_vector_type(2))) float v2f;
typedef __attribute__((ext_vector_type(8))) float v8f;

// ---------------------------------------------------------------------------
// GEMM + bias + ReLU:  out[V,64] = relu(in[V,cin] @ W[cin,64] + b)
// One wave -> 16 rows x 64 cols via 4 WMMA f32 16x16x4 accumulators.
// ---------------------------------------------------------------------------
__global__ __launch_bounds__(128)
void gemm_relu_wmma(const float* __restrict__ in, int ldin, int cin,
                    const float* __restrict__ W,   // [cin, 64] row-major
                    const float* __restrict__ bias,
                    float* __restrict__ out,       // [V, 64]
                    int V)
{
    const int wave = threadIdx.x >> 5;
    const int lane = threadIdx.x & 31;
    const int vbase = (blockIdx.x * 4 + wave) * 16;
    if (vbase >= V) return;                 // wave-uniform; EXEC stays all-1s

    const int m   = lane & 15;              // A row / B-C col within tile
    const int kh  = (lane >> 4) * 2;        // lane-half selects K {0,1} vs {2,3}

    v8f acc0 = {}, acc1 = {}, acc2 = {}, acc3 = {};
    const float* arow = in + (size_t)(vbase + m) * ldin;

    for (int k = 0; k < cin; k += 4) {
        // A fragment (16x4 f32): lane=M, VGPR0/1 = K kh, kh+1
        v2f a;
        a.x = arow[k + kh];
        a.y = arow[k + kh + 1];

        // B fragments (4x16 f32): lane=N, VGPR0/1 = K kh, kh+1
        const float* w0 = W + (size_t)(k + kh) * 64;
        const float* w1 = W + (size_t)(k + kh + 1) * 64;
        v2f b0; b0.x = w0[m];        b0.y = w1[m];
        v2f b1; b1.x = w0[m + 16];   b1.y = w1[m + 16];
        v2f b2; b2.x = w0[m + 32];   b2.y = w1[m + 32];
        v2f b3; b3.x = w0[m + 48];   b3.y = w1[m + 48];

        acc0 = __builtin_amdgcn_wmma_f32_16x16x4_f32(false, a, false, b0, (short)0, acc0, false, false);
        acc1 = __builtin_amdgcn_wmma_f32_16x16x4_f32(false, a, false, b1, (short)0, acc1, false, false);
        acc2 = __builtin_amdgcn_wmma_f32_16x16x4_f32(false, a, false, b2, (short)0, acc2, false, false);
        acc3 = __builtin_amdgcn_wmma_f32_16x16x4_f32(false, a, false, b3, (short)0, acc3, false, false);
    }

    // C/D layout: VGPR r -> M = (lane>>4)*8 + r, N = lane&15
    const int orow = vbase + (lane >> 4) * 8;
    const int ocol = lane & 15;
    const float bb0 = bias[ocol];
    const float bb1 = bias[ocol + 16];
    const float bb2 = bias[ocol + 32];
    const float bb3 = bias[ocol + 48];
#pragma unroll
    for (int r = 0; r < 8; ++r) {
        float* o = out + (size_t)(orow + r) * 64 + ocol;
        o[0]  = fmaxf(acc0[r] + bb0, 0.0f);
        o[16] = fmaxf(acc1[r] + bb1, 0.0f);
        o[32] = fmaxf(acc2[r] + bb2, 0.0f);
        o[48] = fmaxf(acc3[r] + bb3, 0.0f);
    }
}

// ---------------------------------------------------------------------------
// KNN weighted aggregate: one wave handles TWO vertices (one per half-wave).
// Each lane owns 4 features (float4); a half-wave's 16x16B = one full 256B
// neighbor row per k-step -> 512B gathered per wave-step via global_load_b128.
//   w = exp(-10*dsq);  out[:, 0:64)  = mean_k(w*h[idx]) - h[v]
//                      out[:,64:128) = max_k (w*h[idx]) - h[v]
// `out` points at d_out + layer*128, row stride 448 floats.
// ---------------------------------------------------------------------------
__global__ __launch_bounds__(256)
void knn_agg(const float* __restrict__ h,     // [V,64]
             const float* __restrict__ dsq,   // [V,64]
             const int*   __restrict__ idx,   // [V,64]
             float* __restrict__ out,         // base + layer*128, ld = 448
             int V)
{
    const int wave = threadIdx.x >> 5;
    const int lane = threadIdx.x & 31;
    const int vA = (blockIdx.x * 8 + wave) * 2;   // this wave: vertices vA, vA+1
    if (vA >= V) return;

    const int half = lane >> 4;                   // 0 -> vA, 1 -> vA+1
    const int hl   = lane & 15;                   // feature group [hl*4, hl*4+4)
    const int v    = vA + half;

    // Preload 64 indices/weights per vertex, striped so that chunk register
    // j holds k = hl + 16*j for the half-wave's own vertex.
    const size_t vk = (size_t)v * 64 + hl;
    int   ik[4];
    float wk[4];
#pragma unroll
    for (int j = 0; j < 4; ++j) {
        ik[j] = idx[vk + 16 * j];
        wk[j] = __expf(-10.0f * dsq[vk + 16 * j]);
    }

    const int selb4 = (lane & 16) << 2;           // bpermute byte-base per half
    const int foff  = hl << 4;                    // byte offset of my float4 in a row

    float4 s = make_float4(0.f, 0.f, 0.f, 0.f);
    float4 m = make_float4(-INFINITY, -INFINITY, -INFINITY, -INFINITY);

#pragma unroll
    for (int j = 0; j < 4; ++j) {
#pragma unroll
        for (int t = 0; t < 16; ++t) {
            const int bidx = selb4 + (t << 2);
            const int n  = __builtin_amdgcn_ds_bpermute(bidx, ik[j]);
            const float w = __int_as_float(
                __builtin_amdgcn_ds_bpermute(bidx, __float_as_int(wk[j])));
            // gather my float4 of neighbor row n (SADDR=h, VADDR=n*256+foff)
            const float4 f = *(const float4*)((const char*)h + ((size_t)(unsigned)n << 8) + foff);
            const float p0 = w * f.x, p1 = w * f.y, p2 = w * f.z, p3 = w * f.w;
            s.x += p0; s.y += p1; s.z += p2; s.w += p3;
            m.x = fmaxf(m.x, p0); m.y = fmaxf(m.y, p1);
            m.z = fmaxf(m.z, p2); m.w = fmaxf(m.w, p3);
        }
    }

    const float4 self = *(const float4*)((const char*)h + ((size_t)v << 8) + foff);
    float4 mean_o, max_o;
    mean_o.x = s.x * (1.0f / 64.0f) - self.x;
    mean_o.y = s.y * (1.0f / 64.0f) - self.y;
    mean_o.z = s.z * (1.0f / 64.0f) - self.z;
    mean_o.w = s.w * (1.0f / 64.0f) - self.w;
    max_o.x  = m.x - self.x;
    max_o.y  = m.y - self.y;
    max_o.z  = m.z - self.z;
    max_o.w  = m.w - self.w;

    float* orow = out + (size_t)v * 448;
    *(float4*)(orow + (hl << 2))      = mean_o;   // cols [0,64)
    *(float4*)(orow + 64 + (hl << 2)) = max_o;    // cols [64,128)
}

// ---------------------------------------------------------------------------
// Copy x into output columns [384, 448)
// ---------------------------------------------------------------------------
__global__ __launch_bounds__(256)
void copy_x(const float* __restrict__ x, float* __restrict__ out, int V)
{
    int t = blockIdx.x * 256 + threadIdx.x;      // one float4 per thread
    if (t >= V * 16) return;
    int v = t >> 4, c = t & 15;
    ((float4*)(out + (size_t)v * 448 + 384))[c] =
        ((const float4*)(x + (size_t)v * 64))[c];
}

extern "C" void kernel_launch(void* const* d_in, const int* in_sizes, int n_in,
                              void* d_out, int out_size, void* d_ws, size_t ws_size,
                              hipStream_t stream)
{
    const float* x    = (const float*)d_in[0];
    const float* dsq  = (const float*)d_in[1];
    const int*   idx  = (const int*)  d_in[2];
    const float* W0   = (const float*)d_in[3];
    const float* b0   = (const float*)d_in[4];
    const float* W1   = (const float*)d_in[5];
    const float* b1   = (const float*)d_in[6];
    const float* W2   = (const float*)d_in[7];
    const float* b2   = (const float*)d_in[8];
    float* out = (float*)d_out;

    const int V = in_sizes[0] / 64;
    float* h = (float*)d_ws;                       // [V,64] scratch, 25.6 MB

    const int gemm_grid = (V + 63) / 64;           // 4 waves x 16 rows / block
    const int agg_grid  = (V + 15) / 16;           // 8 waves x 2 vertices / block

    copy_x<<<(V * 16 + 255) / 256, 256, 0, stream>>>(x, out, V);

    // Layer 0: in = x [V,64]
    gemm_relu_wmma<<<gemm_grid, 128, 0, stream>>>(x, 64, 64, W0, b0, h, V);
    knn_agg<<<agg_grid, 256, 0, stream>>>(h, dsq, idx, out + 0, V);

    // Layer 1: in = out cols [0,128), ld 448
    gemm_relu_wmma<<<gemm_grid, 128, 0, stream>>>(out + 0, 448, 128, W1, b1, h, V);
    knn_agg<<<agg_grid, 256, 0, stream>>>(h, dsq, idx, out + 128, V);

    // Layer 2: in = out cols [128,256), ld 448
    gemm_relu_wmma<<<gemm_grid, 128, 0, stream>>>(out + 128, 448, 128, W2, b2, h, V);
    knn_agg<<<agg_grid, 256, 0, stream>>>(h, dsq, idx, out + 256, V);
}